// FiveCardClassificationV2_11227044512471
// MI455X (gfx1250) — compile-verified
//
#include <hip/hip_runtime.h>
#include <hip/hip_bf16.h>

typedef float v2f __attribute__((ext_vector_type(2)));
typedef float v8f __attribute__((ext_vector_type(8)));

#define NCH   64
#define HOUT  11
#define WOUTD 10
#define MTOT  4096
#define ROWS_PER_C (MTOT * HOUT)        // 45056 rows per channel
#define TILES_PER_C (ROWS_PER_C / 16)   // 2816 tiles per channel
#define WPB   4                         // waves (16-row tiles) per block
#define FEAT  11264
#define BN_EPS 1e-5f

// workspace layout (floats)
#define WS_SUM0 0
#define WS_SQ0  64
#define WS_SC0  128
#define WS_SH0  192
#define WS_SUM1 256
#define WS_SQ1  320
#define WS_SC1  384
#define WS_SH1  448
#define WS_SUM2 512
#define WS_SQ2  576
#define WS_SC2  640
#define WS_SH2  704
#define WS_FLOATS 768

// ---------------------------------------------------------------------------
// CDNA5 async global->LDS copy (ASYNCcnt-tracked). The LDS destination
// address is the low 32 bits of the generic pointer.
// ---------------------------------------------------------------------------
__device__ __forceinline__ void async_copy_f32(float* lds_dst, const float* gsrc) {
  unsigned lds_off = (unsigned)(unsigned long long)lds_dst;
  asm volatile("global_load_async_to_lds_b32 %0, %1, off"
               :: "v"(lds_off), "v"(gsrc)
               : "memory");
}
__device__ __forceinline__ void async_wait0() {
  asm volatile("s_wait_asynccnt 0x0" ::: "memory");
}

// ---------------------------------------------------------------------------
// init: zero BN accumulators, preload bias into output (chain<3> atomically
// accumulates on top of it)
// ---------------------------------------------------------------------------
__global__ void k_init(float* __restrict__ ws, float* __restrict__ out,
                       const float* __restrict__ b_out) {
  int i = blockIdx.x * 256 + threadIdx.x;
  if (i < WS_FLOATS) ws[i] = 0.f;
  if (i < MTOT * 9) out[i] = b_out[i % 9];
}

// ---------------------------------------------------------------------------
// BN0 statistics: conv output recomputed from one-hot structure
// grid (64, MTOT/128), block 128; thread = one sample m for channel c
// ---------------------------------------------------------------------------
__global__ __launch_bounds__(128)
void k_bn0_stats(const int* __restrict__ xin, const float* __restrict__ conv_w,
                 const float* __restrict__ conv_b, float* __restrict__ ws) {
  __shared__ float cw[128];
  __shared__ float rsum[128];
  __shared__ float rsq[128];
  const int c   = blockIdx.x;
  const int tid = threadIdx.x;
  const int m   = blockIdx.y * 128 + tid;
  for (int i = tid; i < 125; i += 128) async_copy_f32(&cw[i], conv_w + c * 125 + i);
  async_wait0();
  __syncthreads();

  const float cb = conv_b[c];
  const int* xp = xin + m * 10;
  int rk[5], st[5];
#pragma unroll
  for (int i = 0; i < 5; ++i) { rk[i] = xp[i * 2]; st[i] = xp[i * 2 + 1]; }

  float s = 0.f, sq = 0.f;
  for (int h = 0; h < HOUT; ++h) {
    for (int w = 0; w < WOUTD; ++w) {
      float a = cb;
#pragma unroll
      for (int i = 0; i < 5; ++i) {
        int dr = rk[i] - h, ds = st[i] - w;
        if (dr >= 0 && dr < 5 && ds >= 0 && ds < 5) a += cw[(i * 5 + dr) * 5 + ds];
      }
      s += a; sq += a * a;
    }
  }
  rsum[tid] = s; rsq[tid] = sq;
  __syncthreads();
  for (int off = 64; off > 0; off >>= 1) {
    if (tid < off) { rsum[tid] += rsum[tid + off]; rsq[tid] += rsq[tid + off]; }
    __syncthreads();
  }
  if (tid == 0) {
    atomicAdd(&ws[WS_SUM0 + c], rsum[0]);
    atomicAdd(&ws[WS_SQ0 + c], rsq[0]);
  }
}

// ---------------------------------------------------------------------------
// finalize: fused BN scale/shift:  y = x*scale + shift
// ---------------------------------------------------------------------------
__global__ void k_bn_final(const float* __restrict__ sum, const float* __restrict__ sq,
                           const float* __restrict__ g, const float* __restrict__ b,
                           float* __restrict__ scale, float* __restrict__ shift,
                           float invn) {
  int c = threadIdx.x;
  float mean = sum[c] * invn;
  float var  = sq[c] * invn - mean * mean;
  float sc   = g[c] * rsqrtf(var + BN_EPS);
  scale[c] = sc;
  shift[c] = b[c] - mean * sc;
}

// ---------------------------------------------------------------------------
// Fused WMMA chain. One wave = 16 rows (same channel c).
//   STAGE 1: through w_h1 (x2), accumulate BN1 sums
//   STAGE 2: through w_h2 (x3), accumulate BN2 sums
//   STAGE 3: full chain + output projection (atomicAdd into out)
// All weights staged into LDS once per block via async-LDS loads; B-fragment
// reads are unconditional ds loads (w_rank zero-padded to K=12 in LDS).
// A-frag f32 16x4: lane<16 -> K={k0,k0+1}; lane>=16 -> K={k0+2,k0+3}; row=lane%16
// B-frag 4x16:     VGPR v, lane L -> K = k0 + v + 2*(L>=16), N = L%16
// C-frag 16x16:    VGPR g, lane L -> row = g + 8*(L>=16),    col = L%16
// ---------------------------------------------------------------------------
template <int STAGE>
__global__ __launch_bounds__(128)
void k_chain(const int* __restrict__ xin, const float* __restrict__ conv_w,
             const float* __restrict__ conv_b,
             const float* __restrict__ w_rank, const float* __restrict__ b_rank,
             const float* __restrict__ w_h1,   const float* __restrict__ b_h1,
             const float* __restrict__ w_h2,   const float* __restrict__ b_h2,
             const float* __restrict__ w_out,
             float* __restrict__ ws, float* __restrict__ out) {
  __shared__ float cw[128];
  __shared__ float wrkp[32][12];        // zero-padded w_rank (K 10 -> 12)
  __shared__ float wh1p[32][33];
  __shared__ float wh2p[16][33];
  __shared__ float xbuf[WPB][16][33];
  __shared__ float rsum[128];
  __shared__ float rsq[128];

  const int c    = blockIdx.x;
  const int tid  = threadIdx.x;
  const int wave = tid >> 5;
  const int L    = tid & 31;
  const int hsel = L >> 4;        // 0: lanes 0-15, 1: lanes 16-31
  const int lr   = L & 15;

  // ---- stage all weights into LDS (async, ASYNCcnt-tracked) ----
  for (int i = tid; i < 125; i += 128)  async_copy_f32(&cw[i], conv_w + c * 125 + i);
  for (int i = tid; i < 320; i += 128)  async_copy_f32(&wrkp[i / 10][i % 10], w_rank + i);
  for (int i = tid; i < 1024; i += 128) async_copy_f32(&wh1p[i >> 5][i & 31], w_h1 + i);
  for (int i = tid; i < 512; i += 128)  async_copy_f32(&wh2p[i >> 5][i & 31], w_h2 + i);
  if (tid < 64) wrkp[tid >> 1][10 + (tid & 1)] = 0.f;   // zero padding cols
  const float sc0 = ws[WS_SC0 + c], sh0 = ws[WS_SH0 + c];
  async_wait0();
  __syncthreads();

  const int tile    = blockIdx.y * WPB + wave;
  const int rowBase = tile * 16;

  // ---- phase 0: conv + BN0 + relu -> x0[16 rows][10] (padded to 12) ----
  {
    const int row = rowBase + lr;       // row within channel block
    const int m = row / HOUT, h = row % HOUT;
    const int* xp = xin + m * 10;
    int rk[5], st[5];
#pragma unroll
    for (int i = 0; i < 5; ++i) { rk[i] = xp[i * 2]; st[i] = xp[i * 2 + 1]; }
    const float cb = conv_b[c];
    float v[5];
#pragma unroll
    for (int k = 0; k < 5; ++k) {
      const int w = hsel * 5 + k;
      float a = cb;
#pragma unroll
      for (int i = 0; i < 5; ++i) {
        int dr = rk[i] - h, ds = st[i] - w;
        if (dr >= 0 && dr < 5 && ds >= 0 && ds < 5) a += cw[(i * 5 + dr) * 5 + ds];
      }
      v[k] = fmaxf(a * sc0 + sh0, 0.f);
    }
#pragma unroll
    for (int k = 0; k < 5; ++k) xbuf[wave][lr][hsel * 5 + k] = v[k];
    if (hsel == 0) { xbuf[wave][lr][10] = 0.f; xbuf[wave][lr][11] = 0.f; }
  }
  __syncthreads();

  // ---- phase A: x1 = relu(x0 @ w_rank^T + b_rank)   (K=10 pad 12) ----
  v8f acc0 = {}, acc1 = {};
  {
    const float br0 = b_rank[lr], br1 = b_rank[16 + lr];
#pragma unroll
    for (int g = 0; g < 8; ++g) { acc0[g] = br0; acc1[g] = br1; }
#pragma unroll
    for (int ks = 0; ks < 3; ++ks) {
      const int ka = ks * 4 + 2 * hsel;
      v2f a, b0, b1;
      a[0]  = xbuf[wave][lr][ka];
      a[1]  = xbuf[wave][lr][ka + 1];
      b0[0] = wrkp[lr][ka];
      b0[1] = wrkp[lr][ka + 1];
      b1[0] = wrkp[16 + lr][ka];
      b1[1] = wrkp[16 + lr][ka + 1];
      acc0 = __builtin_amdgcn_wmma_f32_16x16x4_f32(false, a, false, b0, (short)0, acc0, false, false);
      acc1 = __builtin_amdgcn_wmma_f32_16x16x4_f32(false, a, false, b1, (short)0, acc1, false, false);
    }
  }
  __syncthreads();
#pragma unroll
  for (int g = 0; g < 8; ++g) {
    xbuf[wave][g + 8 * hsel][lr]      = fmaxf(acc0[g], 0.f);
    xbuf[wave][g + 8 * hsel][16 + lr] = fmaxf(acc1[g], 0.f);
  }
  __syncthreads();

  // ---- phase B: x2 = x1 @ w_h1^T + b_h1   (K=32) ----
  {
    const float bh0 = b_h1[lr], bh1 = b_h1[16 + lr];
#pragma unroll
    for (int g = 0; g < 8; ++g) { acc0[g] = bh0; acc1[g] = bh1; }
#pragma unroll
    for (int ks = 0; ks < 8; ++ks) {
      const int ka = ks * 4 + 2 * hsel;
      v2f a, b0, b1;
      a[0]  = xbuf[wave][lr][ka];
      a[1]  = xbuf[wave][lr][ka + 1];
      b0[0] = wh1p[lr][ka];
      b0[1] = wh1p[lr][ka + 1];
      b1[0] = wh1p[16 + lr][ka];
      b1[1] = wh1p[16 + lr][ka + 1];
      acc0 = __builtin_amdgcn_wmma_f32_16x16x4_f32(false, a, false, b0, (short)0, acc0, false, false);
      acc1 = __builtin_amdgcn_wmma_f32_16x16x4_f32(false, a, false, b1, (short)0, acc1, false, false);
    }
  }

  if constexpr (STAGE == 1) {
    float s = 0.f, sq = 0.f;
#pragma unroll
    for (int g = 0; g < 8; ++g) {
      s  += acc0[g] + acc1[g];
      sq += acc0[g] * acc0[g] + acc1[g] * acc1[g];
    }
    rsum[tid] = s; rsq[tid] = sq;
    __syncthreads();
    for (int off = 64; off > 0; off >>= 1) {
      if (tid < off) { rsum[tid] += rsum[tid + off]; rsq[tid] += rsq[tid + off]; }
      __syncthreads();
    }
    if (tid == 0) {
      atomicAdd(&ws[WS_SUM1 + c], rsum[0]);
      atomicAdd(&ws[WS_SQ1 + c], rsq[0]);
    }
    return;
  } else {
    // y2 = relu(bn1(x2))
    const float sc1 = ws[WS_SC1 + c], sh1 = ws[WS_SH1 + c];
    __syncthreads();
#pragma unroll
    for (int g = 0; g < 8; ++g) {
      xbuf[wave][g + 8 * hsel][lr]      = fmaxf(acc0[g] * sc1 + sh1, 0.f);
      xbuf[wave][g + 8 * hsel][16 + lr] = fmaxf(acc1[g] * sc1 + sh1, 0.f);
    }
    __syncthreads();

    // ---- phase C: x3 = y2 @ w_h2^T + b_h2   (K=32, N=16) ----
    {
      const float bh = b_h2[lr];
#pragma unroll
      for (int g = 0; g < 8; ++g) acc0[g] = bh;
#pragma unroll
      for (int ks = 0; ks < 8; ++ks) {
        const int ka = ks * 4 + 2 * hsel;
        v2f a, b0;
        a[0]  = xbuf[wave][lr][ka];
        a[1]  = xbuf[wave][lr][ka + 1];
        b0[0] = wh2p[lr][ka];
        b0[1] = wh2p[lr][ka + 1];
        acc0 = __builtin_amdgcn_wmma_f32_16x16x4_f32(false, a, false, b0, (short)0, acc0, false, false);
      }
    }

    if constexpr (STAGE == 2) {
      float s = 0.f, sq = 0.f;
#pragma unroll
      for (int g = 0; g < 8; ++g) { s += acc0[g]; sq += acc0[g] * acc0[g]; }
      rsum[tid] = s; rsq[tid] = sq;
      __syncthreads();
      for (int off = 64; off > 0; off >>= 1) {
        if (tid < off) { rsum[tid] += rsum[tid + off]; rsq[tid] += rsq[tid + off]; }
        __syncthreads();
      }
      if (tid == 0) {
        atomicAdd(&ws[WS_SUM2 + c], rsum[0]);
        atomicAdd(&ws[WS_SQ2 + c], rsq[0]);
      }
      return;
    } else {
      // y3 = relu(bn2(x3)); project onto w_out and accumulate into out
      const float sc2 = ws[WS_SC2 + c], sh2 = ws[WS_SH2 + c];
      __syncthreads();
#pragma unroll
      for (int g = 0; g < 8; ++g)
        xbuf[wave][g + 8 * hsel][lr] = fmaxf(acc0[g] * sc2 + sh2, 0.f);
      __syncthreads();

      if (L < 16) {
        const int row = rowBase + L;
        const int m = row / HOUT, h = row % HOUT;
        const float* wo = w_out + (size_t)(c * HOUT + h) * 16;
        float y[16];
#pragma unroll
        for (int j = 0; j < 16; ++j) y[j] = xbuf[wave][L][j];
#pragma unroll
        for (int o = 0; o < 9; ++o) {
          float t = 0.f;
#pragma unroll
          for (int j = 0; j < 16; ++j) t += y[j] * wo[(size_t)o * FEAT + j];
          atomicAdd(&out[m * 9 + o], t);
        }
      }
    }
  }
}

// ---------------------------------------------------------------------------
extern "C" void kernel_launch(void* const* d_in, const int* in_sizes, int n_in,
                              void* d_out, int out_size, void* d_ws, size_t ws_size,
                              hipStream_t stream) {
  const int*   x      = (const int*)d_in[0];
  const float* conv_w = (const float*)d_in[1];
  const float* conv_b = (const float*)d_in[2];
  const float* bn0_g  = (const float*)d_in[3];
  const float* bn0_b  = (const float*)d_in[4];
  const float* w_rank = (const float*)d_in[5];
  const float* b_rank = (const float*)d_in[6];
  const float* w_h1   = (const float*)d_in[7];
  const float* b_h1   = (const float*)d_in[8];
  const float* bn1_g  = (const float*)d_in[9];
  const float* bn1_b  = (const float*)d_in[10];
  const float* w_h2   = (const float*)d_in[11];
  const float* b_h2   = (const float*)d_in[12];
  const float* bn2_g  = (const float*)d_in[13];
  const float* bn2_b  = (const float*)d_in[14];
  const float* w_out  = (const float*)d_in[15];
  const float* b_out  = (const float*)d_in[16];
  float* out = (float*)d_out;
  float* W   = (float*)d_ws;

  k_init<<<(MTOT * 9 + 255) / 256, 256, 0, stream>>>(W, out, b_out);

  k_bn0_stats<<<dim3(NCH, MTOT / 128), 128, 0, stream>>>(x, conv_w, conv_b, W);
  k_bn_final<<<1, 64, 0, stream>>>(W + WS_SUM0, W + WS_SQ0, bn0_g, bn0_b,
                                   W + WS_SC0, W + WS_SH0,
                                   1.f / (float)(MTOT * HOUT * WOUTD));

  dim3 cgrid(NCH, TILES_PER_C / WPB);
  k_chain<1><<<cgrid, 128, 0, stream>>>(x, conv_w, conv_b, w_rank, b_rank,
                                        w_h1, b_h1, w_h2, b_h2, w_out, W, out);
  k_bn_final<<<1, 64, 0, stream>>>(W + WS_SUM1, W + WS_SQ1, bn1_g, bn1_b,
                                   W + WS_SC1, W + WS_SH1,
                                   1.f / (float)(MTOT * HOUT * 32));

  k_chain<2><<<cgrid, 128, 0, stream>>>(x, conv_w, conv_b, w_rank, b_rank,
                                        w_h1, b_h1, w_h2, b_h2, w_out, W, out);
  k_bn_final<<<1, 64, 0, stream>>>(W + WS_SUM2, W + WS_SQ2, bn2_g, bn2_b,
                                   W + WS_SC2, W + WS_SH2,
                                   1.f / (float)(MTOT * HOUT * 16));

  k_chain<3><<<cgrid, 128, 0, stream>>>(x, conv_w, conv_b, w_rank, b_rank,
                                        w_h1, b_h1, w_h2, b_h2, w_out, W, out);
}